// LSTMTagger_22832046146376
// MI455X (gfx1250) — compile-verified
//
#include <hip/hip_runtime.h>
#include <hip/hip_bf16.h>

typedef float v2f __attribute__((ext_vector_type(2)));
typedef float v8f __attribute__((ext_vector_type(8)));

#define Bn   1024   // sentences
#define Sn   128    // words / sentence
#define LPn  14     // padded chars / word
#define Kn   3      // char window
#define CEn  6      // char emb
#define WEn  10     // word emb
#define Lfn  4      // cnn filters
#define Hn   6      // lstm hidden
#define An   100    // alphabet
#define Tn   135    // tagset
#define Dn   14     // WE + Lf
#define LWn  12     // LP - K + 1
#define G4n  24     // 4*H
#define DHn  12     // 2*H

__device__ __forceinline__ float frcp(float x)  { return __builtin_amdgcn_rcpf(x); }
__device__ __forceinline__ float sigf(float x)  { return frcp(1.0f + __expf(-x)); }
__device__ __forceinline__ float tanhq(float x) { return 1.0f - 2.0f * frcp(1.0f + __expf(2.0f * x)); }

// ---------------- Kernel A: embeddings + char CNN -> x[B,S,14] ----------------
__global__ __launch_bounds__(128) void embed_cnn_kernel(
    const int* __restrict__ word_idx, const int* __restrict__ char_idx,
    const float* __restrict__ word_emb, const float* __restrict__ char_emb,
    const float* __restrict__ Wc, const float* __restrict__ bc,
    float* __restrict__ x)
{
    __shared__ float sWc[Lfn * Kn * CEn];   // 72
    __shared__ float sbc[Lfn];
    __shared__ float sce[An * CEn];         // 600 floats, whole char-emb table
    int tid = threadIdx.x;
    for (int i = tid; i < Lfn * Kn * CEn; i += blockDim.x) sWc[i] = Wc[i];
    for (int i = tid; i < Lfn;            i += blockDim.x) sbc[i] = bc[i];
    for (int i = tid; i < An * CEn;       i += blockDim.x) sce[i] = char_emb[i];
    __syncthreads();

    int gid = blockIdx.x * blockDim.x + tid;   // word index in [0, B*S)
    if (gid >= Bn * Sn) return;

    int cidx[LPn];
    const int* cp = char_idx + gid * LPn;
#pragma unroll
    for (int p = 0; p < LPn; ++p) cidx[p] = cp[p];

    float mx0 = -1e30f, mx1 = -1e30f, mx2 = -1e30f, mx3 = -1e30f;
#pragma unroll
    for (int w = 0; w < LWn; ++w) {
        float a0 = sbc[0], a1 = sbc[1], a2 = sbc[2], a3 = sbc[3];
#pragma unroll
        for (int c = 0; c < Kn; ++c) {
            const float* ce = &sce[cidx[w + c] * CEn];
#pragma unroll
            for (int e = 0; e < CEn; ++e) {
                float v = ce[e];
                int  k = c * CEn + e;
                a0 = fmaf(sWc[0 * 18 + k], v, a0);
                a1 = fmaf(sWc[1 * 18 + k], v, a1);
                a2 = fmaf(sWc[2 * 18 + k], v, a2);
                a3 = fmaf(sWc[3 * 18 + k], v, a3);
            }
        }
        mx0 = fmaxf(mx0, a0); mx1 = fmaxf(mx1, a1);
        mx2 = fmaxf(mx2, a2); mx3 = fmaxf(mx3, a3);
    }

    float* xo = x + (size_t)gid * Dn;
    const float* we = word_emb + (size_t)word_idx[gid] * WEn;
#pragma unroll
    for (int e = 0; e < WEn; ++e) xo[e] = we[e];
    xo[10] = mx0; xo[11] = mx1; xo[12] = mx2; xo[13] = mx3;
}

// ------------- Kernel B: bidirectional LSTM scan with f32 WMMA ---------------
// grid = (B/16, 2); 1 wave per block; each wave owns a 16-row batch tile.
// Gates G[16,24] = [x_t ; h] (16x20) @ [Wih^T ; Whh^T] (20x24) + b, done as
// 5 K-chunks x 2 N-tiles of V_WMMA_F32_16X16X4_F32 with bias preloaded in C.
__global__ __launch_bounds__(32) void bilstm_kernel(
    const float* __restrict__ x,
    const float* __restrict__ Wih_f, const float* __restrict__ Whh_f, const float* __restrict__ b_f,
    const float* __restrict__ Wih_b, const float* __restrict__ Whh_b, const float* __restrict__ b_b,
    float* __restrict__ hcat)
{
    const int mtile = blockIdx.x;     // 0..63
    const int dir   = blockIdx.y;     // 0 = fwd, 1 = bwd
    const float* Wih = dir ? Wih_b : Wih_f;   // [24,14]
    const float* Whh = dir ? Whh_b : Whh_f;   // [24, 6]
    const float* bg  = dir ? b_b   : b_f;     // [24]

    const int lane  = threadIdx.x;
    const int lrow  = lane & 15;      // A: M row / B,D: N col
    const int khalf = lane >> 4;      // 0: K even pair, 1: K odd pair
    const int grow  = mtile * 16 + lrow;

    // Loop-invariant B matrix (K=20 x N=24, zero-padded N->32), WMMA B layout.
    v2f Bw[5][2];
#pragma unroll
    for (int q = 0; q < 5; ++q) {
#pragma unroll
        for (int nt = 0; nt < 2; ++nt) {
            int n  = nt * 16 + lrow;
            int k0 = q * 4 + khalf * 2;
            float w0 = 0.f, w1 = 0.f;
            if (n < G4n) {
                w0 = (k0     < Dn) ? Wih[n * Dn + k0]       : Whh[n * Hn + (k0 - Dn)];
                w1 = (k0 + 1 < Dn) ? Wih[n * Dn + k0 + 1]   : Whh[n * Hn + (k0 + 1 - Dn)];
            }
            Bw[q][nt].x = w0; Bw[q][nt].y = w1;
        }
    }
    float b0 = bg[lrow];
    float b1 = (16 + lrow) < G4n ? bg[16 + lrow] : 0.f;
    v8f cb0 = {b0, b0, b0, b0, b0, b0, b0, b0};
    v8f cb1 = {b1, b1, b1, b1, b1, b1, b1, b1};

    __shared__ float hbuf[16][Hn];   // rows 24B -> even idx is 8B-aligned
    __shared__ float cbuf[16][Hn];
    __shared__ float G[16][32];
    for (int i = lane; i < 16 * Hn; i += 32) { (&hbuf[0][0])[i] = 0.f; (&cbuf[0][0])[i] = 0.f; }
    __syncthreads();

    for (int t = 0; t < Sn; ++t) {
        int tt = dir ? (Sn - 1 - t) : t;
        const float* xr = x + ((size_t)grow * Sn + tt) * Dn;   // 56B stride -> 8B aligned

        // A tile 16x20: K 0..13 from x_t, K 14..19 from h.
        // q=0..2 always x; q=3 splits by khalf; q=4 always h. All v2f (b64) loads.
        v2f Aq[5];
#pragma unroll
        for (int q = 0; q < 3; ++q)
            Aq[q] = *(const v2f*)(xr + q * 4 + khalf * 2);
        if (khalf == 0) Aq[3] = *(const v2f*)(xr + 12);
        else            Aq[3] = *(const v2f*)(&hbuf[lrow][0]);
        Aq[4] = *(const v2f*)(&hbuf[lrow][2 + khalf * 2]);

        v8f d0 = cb0, d1 = cb1;
#pragma unroll
        for (int q = 0; q < 5; ++q) {
            d0 = __builtin_amdgcn_wmma_f32_16x16x4_f32(false, Aq[q], false, Bw[q][0], (short)0, d0, false, false);
            d1 = __builtin_amdgcn_wmma_f32_16x16x4_f32(false, Aq[q], false, Bw[q][1], (short)0, d1, false, false);
        }
#pragma unroll
        for (int v = 0; v < 8; ++v) {
            G[khalf * 8 + v][lrow]      = d0[v];
            G[khalf * 8 + v][16 + lrow] = d1[v];
        }
        __syncthreads();

        // Element-wise LSTM cell: 16 rows x 6 hidden = 96 items over 32 lanes
#pragma unroll
        for (int p = 0; p < 3; ++p) {
            int idx = lane + p * 32;
            int row = idx / Hn;
            int j   = idx % Hn;
            float gi = G[row][j];
            float gf = G[row][j + 6];
            float gg = G[row][j + 12];
            float go = G[row][j + 18];
            float cn = sigf(gf) * cbuf[row][j] + sigf(gi) * tanhq(gg);
            float hn = sigf(go) * tanhq(cn);
            cbuf[row][j] = cn;
            hbuf[row][j] = hn;
            int gr = mtile * 16 + row;
            hcat[((size_t)gr * Sn + tt) * DHn + dir * Hn + j] = hn;
        }
        __syncthreads();
    }
}

// ---------- Kernel C: tag projection (WMMA) + log_softmax -> out[B,S,135] ----
__global__ __launch_bounds__(32) void tag_kernel(
    const float* __restrict__ hcat, const float* __restrict__ Wt,
    const float* __restrict__ bt, float* __restrict__ out)
{
    const int lane  = threadIdx.x;
    const int lrow  = lane & 15;
    const int khalf = lane >> 4;
    const int mrow0 = blockIdx.x * 16;

    // A tile 16x12 from hcat (rows 48B -> even idx 8B-aligned)
    const float* hr = hcat + (size_t)(mrow0 + lrow) * DHn;
    v2f Aq[3];
#pragma unroll
    for (int q = 0; q < 3; ++q)
        Aq[q] = *(const v2f*)(hr + q * 4 + khalf * 2);

    __shared__ float Tg[16][144];
    __shared__ float rmx[16], rls[16];

#pragma unroll
    for (int nt = 0; nt < 9; ++nt) {
        int n = nt * 16 + lrow;
        bool valid = (n < Tn);
        float bv = valid ? bt[n] : 0.f;
        v8f d = {bv, bv, bv, bv, bv, bv, bv, bv};
        const float* wr = Wt + n * DHn;     // rows 48B -> even idx 8B-aligned
#pragma unroll
        for (int q = 0; q < 3; ++q) {
            v2f Bq = {0.f, 0.f};
            if (valid) Bq = *(const v2f*)(wr + q * 4 + khalf * 2);
            d = __builtin_amdgcn_wmma_f32_16x16x4_f32(false, Aq[q], false, Bq, (short)0, d, false, false);
        }
#pragma unroll
        for (int v = 0; v < 8; ++v) Tg[khalf * 8 + v][n] = d[v];
    }
    __syncthreads();

    if (lane < 16) {
        float mx = -1e30f;
        for (int n = 0; n < Tn; ++n) mx = fmaxf(mx, Tg[lane][n]);
        float s = 0.f;
        for (int n = 0; n < Tn; ++n) s += __expf(Tg[lane][n] - mx);
        rmx[lane] = mx;
        rls[lane] = logf(s);
    }
    __syncthreads();

    for (int idx = lane; idx < 16 * Tn; idx += 32) {
        int row = idx / Tn;
        int n   = idx % Tn;
        out[(size_t)(mrow0 + row) * Tn + n] = Tg[row][n] - rmx[row] - rls[row];
    }
}

extern "C" void kernel_launch(void* const* d_in, const int* in_sizes, int n_in,
                              void* d_out, int out_size, void* d_ws, size_t ws_size,
                              hipStream_t stream) {
    (void)in_sizes; (void)n_in; (void)out_size; (void)ws_size;
    const int*   word_idx = (const int*)d_in[0];
    const int*   char_idx = (const int*)d_in[1];
    const float* word_emb = (const float*)d_in[2];
    const float* char_emb = (const float*)d_in[3];
    const float* Wc    = (const float*)d_in[4];
    const float* bc    = (const float*)d_in[5];
    const float* Wih_f = (const float*)d_in[6];
    const float* Whh_f = (const float*)d_in[7];
    const float* b_f   = (const float*)d_in[8];
    const float* Wih_b = (const float*)d_in[9];
    const float* Whh_b = (const float*)d_in[10];
    const float* b_b   = (const float*)d_in[11];
    const float* Wt    = (const float*)d_in[12];
    const float* bt    = (const float*)d_in[13];
    float* outp = (float*)d_out;

    float* xbuf = (float*)d_ws;                         // B*S*14 floats (7.3 MB)
    float* hcat = xbuf + (size_t)Bn * Sn * Dn;          // B*S*12 floats (6.3 MB)

    embed_cnn_kernel<<<(Bn * Sn) / 128, 128, 0, stream>>>(
        word_idx, char_idx, word_emb, char_emb, Wc, bc, xbuf);
    bilstm_kernel<<<dim3(Bn / 16, 2), 32, 0, stream>>>(
        xbuf, Wih_f, Whh_f, b_f, Wih_b, Whh_b, b_b, hcat);
    tag_kernel<<<(Bn * Sn) / 16, 32, 0, stream>>>(hcat, Wt, bt, outp);
}